// GroupShuffleNorm2d_5540507812235
// MI455X (gfx1250) — compile-verified
//
#include <hip/hip_runtime.h>
#include <hip/hip_bf16.h>

typedef int   v4i __attribute__((ext_vector_type(4)));
typedef float v2f __attribute__((ext_vector_type(2)));
typedef float v4f __attribute__((ext_vector_type(4)));
typedef float v8f __attribute__((ext_vector_type(8)));

#if defined(__has_builtin)
#if __has_builtin(__builtin_amdgcn_global_load_async_to_lds_b128)
#define HAVE_ASYNC_LDS 1
#endif
#if __has_builtin(__builtin_amdgcn_wmma_f32_16x16x4_f32)
#define HAVE_WMMA 1
#endif
#endif

namespace {
constexpr int Bn = 32, Cn = 64, Hn = 128, Wn = 128, Gn = 8;
constexpr int HWn = Hn * Wn;                 // 16384 elements per plane
constexpr float EPSf = 1e-5f;
constexpr int THREADS = 256;                 // 8 waves (wave32)
constexpr int TILE_F4 = THREADS;             // 256 float4 = 4 KB per tile
constexpr int NTILES = HWn / (TILE_F4 * 4);  // 16 tiles per plane
constexpr int CHUNKS = HWn / (THREADS * 4);  // 16 chunks per plane (norm pass)
}

// ---------------------------------------------------------------------------
// Kernel 1: per-(b,c) plane sums (s, ss) via async global->LDS double buffer.
// One block per plane. Each thread async-loads the float4 it will consume, so
// only the per-wave in-order ASYNCcnt wait is needed inside the pipeline.
// ---------------------------------------------------------------------------
__global__ __launch_bounds__(THREADS) void gsn_stats(const float* __restrict__ x,
                                                     float2* __restrict__ plane_stats) {
  __shared__ float4 lbuf[2][TILE_F4];
  __shared__ float red[2 * THREADS];

  const int t = threadIdx.x;
  const long long plane = blockIdx.x;
  const float* base = x + plane * (long long)HWn;

  float s = 0.f, ss = 0.f;

#if HAVE_ASYNC_LDS
  // Prime the pipeline: tile 0 -> buffer 0.
  __builtin_amdgcn_global_load_async_to_lds_b128(
      (v4i*)(base + t * 4),
      (v4i*)&lbuf[0][t], 0, 0);

  for (int i = 0; i < NTILES; ++i) {
    if (i + 1 < NTILES) {
      // Prefetch next tile into the other buffer, then wait for tile i only.
      __builtin_amdgcn_global_load_async_to_lds_b128(
          (v4i*)(base + (long long)(i + 1) * (TILE_F4 * 4) + t * 4),
          (v4i*)&lbuf[(i + 1) & 1][t], 0, 0);
      asm volatile("s_wait_asynccnt 1" ::: "memory");
    } else {
      asm volatile("s_wait_asynccnt 0" ::: "memory");
    }
    float4 v = lbuf[i & 1][t];
    s  += v.x + v.y + v.z + v.w;
    ss += v.x * v.x + v.y * v.y + v.z * v.z + v.w * v.w;
  }
#else
  for (int i = 0; i < NTILES; ++i) {
    float4 v = *(const float4*)(base + (long long)i * (TILE_F4 * 4) + t * 4);
    s  += v.x + v.y + v.z + v.w;
    ss += v.x * v.x + v.y * v.y + v.z * v.z + v.w * v.w;
  }
#endif

  // Deterministic block tree reduction.
  red[t] = s;
  red[t + THREADS] = ss;
  __syncthreads();
  for (int stride = THREADS / 2; stride > 0; stride >>= 1) {
    if (t < stride) {
      red[t] += red[t + stride];
      red[t + THREADS] += red[t + THREADS + stride];
    }
    __syncthreads();
  }
  if (t == 0) {
    plane_stats[plane] = make_float2(red[0], red[THREADS]);
  }
}

// ---------------------------------------------------------------------------
// Kernel 2: shuffled segment-sum as a one-hot GEMM on the WMMA pipe.
//   S[b,g]  = sum_c s[b,c]  * M[c,g],  M[c,g] = (gids[c]==g)
//   SS[b,g] = sum_c ss[b,c] * M[c,g]
// [32 x 64] x [64 x 16] f32 GEMM via V_WMMA_F32_16X16X4_F32:
// 2 waves (one per 16-row batch tile), 16 K-steps, 2 accumulator chains.
// Operand layouts per CDNA5 ISA 7.12.2:
//   A 16x4 : VGPR0 = K0(lo lanes)/K2(hi lanes), VGPR1 = K1/K3
//   B 4x16 : VGPR0 = row0(lo)/row2(hi),          VGPR1 = row1/row3, N = lane&15
//   D 16x16: VGPR r = M=r (lo lanes) / M=r+8 (hi lanes), N = lane&15
// ---------------------------------------------------------------------------
#if HAVE_WMMA
__global__ __launch_bounds__(64) void gsn_group_wmma(const float2* __restrict__ plane_stats,
                                                     const int* __restrict__ gids,
                                                     float2* __restrict__ gstats) {
  __shared__ int sg[Cn];
  const int t = threadIdx.x;     // 0..63 -> 2 full waves
  const int lane = t & 31;
  const int bt = t >> 5;         // wave id == batch tile (b = bt*16 .. bt*16+15)
  if (t < Cn) sg[t] = gids[t];
  __syncthreads();

  const int m = lane & 15;       // A row within tile / D column (group id)
  const int hi = lane >> 4;      // lane half selects K pair / D row half
  const int row = bt * 16 + m;   // batch index feeding A

  v8f accS = {};                 // D accumulator for sums
  v8f accQ = {};                 // D accumulator for sum-of-squares

#pragma unroll
  for (int kk = 0; kk < Cn / 4; ++kk) {
    const int c0 = kk * 4 + hi * 2;          // this lane's two K channels
    const float2 p0 = plane_stats[row * Cn + c0];
    const float2 p1 = plane_stats[row * Cn + c0 + 1];
    const int g0 = sg[c0];
    const int g1 = sg[c0 + 1];
    v2f aS = {p0.x, p1.x};
    v2f aQ = {p0.y, p1.y};
    v2f bm = {(g0 == m) ? 1.0f : 0.0f, (g1 == m) ? 1.0f : 0.0f};
    accS = __builtin_amdgcn_wmma_f32_16x16x4_f32(false, aS, false, bm,
                                                 (short)0, accS, false, false);
    accQ = __builtin_amdgcn_wmma_f32_16x16x4_f32(false, aQ, false, bm,
                                                 (short)0, accQ, false, false);
  }

  // Lane holds D[r + hi*8][m] for r=0..7; columns m>=G are zero padding.
  if (m < Gn) {
    int cnt = 0;
    for (int c = 0; c < Cn; ++c) cnt += (sg[c] == m) ? 1 : 0;
    const float n = (float)cnt * (float)HWn;
#pragma unroll
    for (int r = 0; r < 8; ++r) {
      const int b = bt * 16 + hi * 8 + r;
      const float s = accS[r];
      const float q = accQ[r];
      const float mean = s / n;
      const float var = (q - n * mean * mean) / (n - 1.0f);  // unbiased (ddof=1)
      const float inv = rsqrtf(var + EPSf);
      gstats[b * Gn + m] = make_float2(mean, inv);
    }
  }
}
#else
__global__ __launch_bounds__(THREADS) void gsn_group(const float2* __restrict__ plane_stats,
                                                     const int* __restrict__ gids,
                                                     float2* __restrict__ gstats) {
  const int idx = threadIdx.x;
  if (idx >= Bn * Gn) return;
  const int b = idx / Gn;
  const int g = idx % Gn;
  float s = 0.f, ss = 0.f;
  int cnt = 0;
  for (int c = 0; c < Cn; ++c) {
    if (gids[c] == g) {
      float2 p = plane_stats[b * Cn + c];
      s += p.x;
      ss += p.y;
      ++cnt;
    }
  }
  const float n = (float)cnt * (float)HWn;
  const float mean = s / n;
  const float var = (ss - n * mean * mean) / (n - 1.0f);
  const float inv = rsqrtf(var + EPSf);
  gstats[idx] = make_float2(mean, inv);
}
#endif

// ---------------------------------------------------------------------------
// Kernel 3: streaming normalize. Channel is uniform per block, so the group
// map / stats / gamma / beta collapse to scalar loads; one float4 in, one out.
// Output stores are non-temporal so they don't evict x from L2 (192 MB) —
// x's second read should stay L2-resident.
// ---------------------------------------------------------------------------
__global__ __launch_bounds__(THREADS) void gsn_norm(const float* __restrict__ x,
                                                    const float* __restrict__ gamma,
                                                    const float* __restrict__ beta,
                                                    const int* __restrict__ gids,
                                                    const float2* __restrict__ gstats,
                                                    float* __restrict__ out) {
  const int plane = blockIdx.x / CHUNKS;   // b * C + c
  const int chunk = blockIdx.x % CHUNKS;
  const int c = plane % Cn;
  const int b = plane / Cn;
  const int g = gids[c];
  const float2 mi = gstats[b * Gn + g];    // (mean, inv)
  const float scale = mi.y * gamma[c];
  const float bias = beta[c] - mi.x * scale;

  const long long off =
      (long long)plane * HWn + (long long)chunk * (THREADS * 4) + threadIdx.x * 4;
  v4f v = *(const v4f*)(x + off);
  v4f r;
  r.x = v.x * scale + bias;
  r.y = v.y * scale + bias;
  r.z = v.z * scale + bias;
  r.w = v.w * scale + bias;
  __builtin_nontemporal_store(r, (v4f*)(out + off));
}

// ---------------------------------------------------------------------------
extern "C" void kernel_launch(void* const* d_in, const int* in_sizes, int n_in,
                              void* d_out, int out_size, void* d_ws, size_t ws_size,
                              hipStream_t stream) {
  const float* x     = (const float*)d_in[0];
  const float* gamma = (const float*)d_in[1];
  const float* beta  = (const float*)d_in[2];
  const int*   gids  = (const int*)d_in[3];

  float2* plane_stats = (float2*)d_ws;                 // B*C float2 = 16 KB
  float2* gstats      = plane_stats + (size_t)Bn * Cn; // B*G float2 =  2 KB

  gsn_stats<<<Bn * Cn, THREADS, 0, stream>>>(x, plane_stats);
#if HAVE_WMMA
  gsn_group_wmma<<<1, 64, 0, stream>>>(plane_stats, gids, gstats);
#else
  gsn_group<<<1, THREADS, 0, stream>>>(plane_stats, gids, gstats);
#endif
  gsn_norm<<<Bn * Cn * CHUNKS, THREADS, 0, stream>>>(x, gamma, beta, gids, gstats,
                                                     (float*)d_out);
}